// Net_84396107366807
// MI455X (gfx1250) — compile-verified
//
#include <hip/hip_runtime.h>
#include <hip/hip_bf16.h>

typedef __attribute__((ext_vector_type(2))) float v2f;
typedef __attribute__((ext_vector_type(8))) float v8f;

#define N_NODES 100000
#define N_EDGES 1200000
#define EE (N_EDGES + N_NODES)   // edges + self loops

// ---------------------------------------------------------------------------
// Degree / normalization precompute
// ---------------------------------------------------------------------------
__global__ void zero_f32_kernel(float* __restrict__ p, int n) {
    int i = blockIdx.x * blockDim.x + threadIdx.x;
    if (i < n) p[i] = 0.0f;
}

__global__ void accum_deg_kernel(const long long* __restrict__ ei,
                                 const float* __restrict__ ea,
                                 float* __restrict__ deg) {
    int i = blockIdx.x * blockDim.x + threadIdx.x;
    if (i >= EE) return;
    if (i < N_EDGES) {
        int d = (int)ei[N_EDGES + i];            // dst row of edge_index
        unsafeAtomicAdd(&deg[d], ea[i]);
    } else {
        unsafeAtomicAdd(&deg[i - N_EDGES], 1.0f); // self loop weight 1
    }
}

__global__ void deg_to_dis_kernel(float* __restrict__ deg) {
    int i = blockIdx.x * blockDim.x + threadIdx.x;
    if (i >= N_NODES) return;
    float d = deg[i];
    deg[i] = (d > 0.0f) ? rsqrtf(fmaxf(d, 1e-12f)) : 0.0f;
}

__global__ void compute_norm_kernel(const long long* __restrict__ ei,
                                    const float* __restrict__ ea,
                                    const float* __restrict__ dis,
                                    float* __restrict__ nrm) {
    int i = blockIdx.x * blockDim.x + threadIdx.x;
    if (i >= EE) return;
    if (i < N_EDGES) {
        int s = (int)ei[i];
        int d = (int)ei[N_EDGES + i];
        nrm[i] = dis[s] * ea[i] * dis[d];
    } else {
        int n = i - N_EDGES;
        nrm[i] = dis[n] * dis[n];
    }
}

// Pad W4 (64x40) -> (64x48) and b4 (40) -> (48) with zeros.
__global__ void pad_w4_kernel(const float* __restrict__ W4,
                              const float* __restrict__ b4,
                              float* __restrict__ W4p,
                              float* __restrict__ b4p) {
    int i = blockIdx.x * blockDim.x + threadIdx.x;
    if (i < 64 * 48) {
        int k = i / 48, c = i % 48;
        W4p[i] = (c < 40) ? W4[k * 40 + c] : 0.0f;
    } else if (i < 64 * 48 + 48) {
        int c = i - 64 * 48;
        b4p[c] = (c < 40) ? b4[c] : 0.0f;
    }
}

// ---------------------------------------------------------------------------
// Dense transform: Out[N x MOUT] = elu?(A)[N x 64] @ W[64 x MOUT]
// Block = 5 waves (160 thr). W staged into LDS via GLOBAL_LOAD_ASYNC_TO_LDS
// in a pair-interleaved layout ( [k/2][col][k&1] ) so each B fragment is a
// single aligned ds_load_b64. Each wave owns a 16-row strip and computes all
// MOUT columns: MOUT/16 accumulators x 16 K-steps of V_WMMA_F32_16X16X4_F32.
// Branchless ELU keeps EXEC all-ones around the WMMAs.
// ---------------------------------------------------------------------------
template <int MOUT, bool ELU>
__global__ __launch_bounds__(160)
void gemm_wmma_f32_kernel(const float* __restrict__ A,
                          const float* __restrict__ W,
                          float* __restrict__ Out) {
    constexpr int NT = MOUT / 16;
    __shared__ float ldsW[64 * MOUT];   // pair-interleaved: ((k>>1)*MOUT + c)*2 + (k&1)

    const int tid = threadIdx.x;
    for (int i = tid; i < 64 * MOUT; i += 160) {
        int k = i / MOUT, c = i % MOUT;
        unsigned dst = (unsigned)(uintptr_t)&ldsW[((((k >> 1) * MOUT) + c) << 1) | (k & 1)];
        unsigned long long src = (unsigned long long)(const void*)(W + i);
        // async DMA: memory -> LDS, per-lane scattered LDS dest (ASYNCcnt)
        asm volatile("global_load_async_to_lds_b32 %0, %1, off"
                     :: "v"(dst), "v"(src) : "memory");
    }
    asm volatile("s_wait_asynccnt 0x0" ::: "memory");
    __syncthreads();

    const int wave = tid >> 5;            // 0..4
    const int lane = tid & 31;
    const int half = lane >> 4;           // 0 or 1
    const int l    = lane & 15;
    const int m0   = blockIdx.x * 80 + wave * 16;
    const int koff = half * 2;            // lanes 16-31 hold K+2,K+3

    const float* Arow = A + (size_t)(m0 + l) * 64 + koff;

    v8f acc[NT];
#pragma unroll
    for (int t = 0; t < NT; ++t) acc[t] = {};

#pragma unroll
    for (int kk = 0; kk < 64; kk += 4) {
        v2f a = *reinterpret_cast<const v2f*>(Arow + kk);   // global_load_b64
        if (ELU) {
            float e0 = __expf(fminf(a.x, 0.0f)) - 1.0f;     // unconditional exp
            float e1 = __expf(fminf(a.y, 0.0f)) - 1.0f;     // -> cndmask only
            a.x = (a.x > 0.0f) ? a.x : e0;
            a.y = (a.y > 0.0f) ? a.y : e1;
        }
        const int p = (kk + koff) >> 1;                     // K pair index
#pragma unroll
        for (int t = 0; t < NT; ++t) {
            // contiguous pair {W[kab][c], W[kab+1][c]} -> one ds_load_b64
            v2f b = *reinterpret_cast<const v2f*>(&ldsW[(p * MOUT + t * 16 + l) << 1]);
            acc[t] = __builtin_amdgcn_wmma_f32_16x16x4_f32(
                         /*neg_a=*/false, a, /*neg_b=*/false, b,
                         /*c_mod=*/(short)0, acc[t],
                         /*reuse_a=*/false, /*reuse_b=*/false);
        }
    }

#pragma unroll
    for (int t = 0; t < NT; ++t)
#pragma unroll
        for (int r = 0; r < 8; ++r)
            Out[(size_t)(m0 + r + half * 8) * MOUT + t * 16 + l] = acc[t][r];
}

// ---------------------------------------------------------------------------
// Scatter phase: out[dst] += norm[e] * tmp[src]  (E real edges + N self loops)
// One thread per (edge, 4-feature group); float4 gather + 4 fp32 atomics.
// ---------------------------------------------------------------------------
template <int STRIDE>
__global__ void init_bias_kernel(float* __restrict__ outb,
                                 const float* __restrict__ b) {
    int i = blockIdx.x * blockDim.x + threadIdx.x;
    if (i < N_NODES * STRIDE) outb[i] = b[i % STRIDE];
}

template <int STRIDE>
__global__ void scatter_edges_kernel(const float* __restrict__ tmp,
                                     const float* __restrict__ nrm,
                                     const long long* __restrict__ ei,
                                     float* __restrict__ outb) {
    constexpr int G = STRIDE / 4;
    int gid = blockIdx.x * blockDim.x + threadIdx.x;
    if (gid >= EE * G) return;
    int e = gid / G;
    int g = gid % G;
    int s, d;
    if (e < N_EDGES) { s = (int)ei[e]; d = (int)ei[N_EDGES + e]; }
    else             { s = d = e - N_EDGES; }
    float w = nrm[e];
    const float4 v = *reinterpret_cast<const float4*>(tmp + (size_t)s * STRIDE + g * 4);
    float* o = outb + (size_t)d * STRIDE + g * 4;
    unsafeAtomicAdd(o + 0, v.x * w);
    unsafeAtomicAdd(o + 1, v.y * w);
    unsafeAtomicAdd(o + 2, v.z * w);
    unsafeAtomicAdd(o + 3, v.w * w);
}

// ---------------------------------------------------------------------------
// Final log_softmax over 40 classes (input stride 48, bias already included).
// ---------------------------------------------------------------------------
__global__ void log_softmax_kernel(const float* __restrict__ agg,
                                   float* __restrict__ out) {
    int n = blockIdx.x * blockDim.x + threadIdx.x;
    if (n >= N_NODES) return;
    float v[40];
    float m = -3.4e38f;
#pragma unroll
    for (int c = 0; c < 40; ++c) {
        v[c] = agg[(size_t)n * 48 + c];
        m = fmaxf(m, v[c]);
    }
    float s = 0.0f;
#pragma unroll
    for (int c = 0; c < 40; ++c) s += __expf(v[c] - m);
    float lse = m + __logf(s);
#pragma unroll
    for (int c = 0; c < 40; ++c) out[(size_t)n * 40 + c] = v[c] - lse;
}

// ---------------------------------------------------------------------------
extern "C" void kernel_launch(void* const* d_in, const int* in_sizes, int n_in,
                              void* d_out, int out_size, void* d_ws, size_t ws_size,
                              hipStream_t stream) {
    const float*     x  = (const float*)d_in[0];
    const long long* ei = (const long long*)d_in[1];   // int64 [2, E]
    const float*     ea = (const float*)d_in[2];
    const float* W1 = (const float*)d_in[3]; const float* b1 = (const float*)d_in[4];
    const float* W2 = (const float*)d_in[5]; const float* b2 = (const float*)d_in[6];
    const float* W3 = (const float*)d_in[7]; const float* b3 = (const float*)d_in[8];
    const float* W4 = (const float*)d_in[9]; const float* b4 = (const float*)d_in[10];
    float* out = (float*)d_out;

    // Workspace layout (floats)
    float* ws   = (float*)d_ws;
    float* dis  = ws;                                  // N   (deg -> dis)
    float* nrm  = dis + N_NODES;                       // E+N
    float* bufA = nrm + EE;                            // N*64
    float* bufB = bufA + (size_t)N_NODES * 64;         // N*64
    float* bufT = bufB + (size_t)N_NODES * 64;         // N*64 (GEMM temp)
    float* W4p  = bufT + (size_t)N_NODES * 64;         // 64*48
    float* b4p  = W4p + 64 * 48;                       // 48

    const int TB = 256;
    auto blk = [](long long n, int tb) { return (int)((n + tb - 1) / tb); };

    // --- normalization precompute ---
    zero_f32_kernel<<<blk(N_NODES, TB), TB, 0, stream>>>(dis, N_NODES);
    accum_deg_kernel<<<blk(EE, TB), TB, 0, stream>>>(ei, ea, dis);
    deg_to_dis_kernel<<<blk(N_NODES, TB), TB, 0, stream>>>(dis);
    compute_norm_kernel<<<blk(EE, TB), TB, 0, stream>>>(ei, ea, dis, nrm);
    pad_w4_kernel<<<blk(64 * 48 + 48, TB), TB, 0, stream>>>(W4, b4, W4p, b4p);

    const int GEMM_BLOCKS = N_NODES / 80;   // 1250, 5 waves x 16 rows each

    // --- layer 1: x -> bufA ---
    gemm_wmma_f32_kernel<64, false><<<GEMM_BLOCKS, 160, 0, stream>>>(x, W1, bufT);
    init_bias_kernel<64><<<blk((long long)N_NODES * 64, TB), TB, 0, stream>>>(bufA, b1);
    scatter_edges_kernel<64><<<blk((long long)EE * 16, TB), TB, 0, stream>>>(bufT, nrm, ei, bufA);

    // --- layer 2: elu(bufA) -> bufB ---
    gemm_wmma_f32_kernel<64, true><<<GEMM_BLOCKS, 160, 0, stream>>>(bufA, W2, bufT);
    init_bias_kernel<64><<<blk((long long)N_NODES * 64, TB), TB, 0, stream>>>(bufB, b2);
    scatter_edges_kernel<64><<<blk((long long)EE * 16, TB), TB, 0, stream>>>(bufT, nrm, ei, bufB);

    // --- layer 3: elu(bufB) -> bufA ---
    gemm_wmma_f32_kernel<64, true><<<GEMM_BLOCKS, 160, 0, stream>>>(bufB, W3, bufT);
    init_bias_kernel<64><<<blk((long long)N_NODES * 64, TB), TB, 0, stream>>>(bufA, b3);
    scatter_edges_kernel<64><<<blk((long long)EE * 16, TB), TB, 0, stream>>>(bufT, nrm, ei, bufA);

    // --- layer 4 (padded to 48 cols): elu(bufA) -> bufB ---
    gemm_wmma_f32_kernel<48, true><<<GEMM_BLOCKS, 160, 0, stream>>>(bufA, W4p, bufT);
    init_bias_kernel<48><<<blk((long long)N_NODES * 48, TB), TB, 0, stream>>>(bufB, b4p);
    scatter_edges_kernel<48><<<blk((long long)EE * 12, TB), TB, 0, stream>>>(bufT, nrm, ei, bufB);

    // --- log_softmax over 40 classes ---
    log_softmax_kernel<<<blk(N_NODES, TB), TB, 0, stream>>>(bufB, out);
}